// ClusterLoss_82317343195278
// MI455X (gfx1250) — compile-verified
//
#include <hip/hip_runtime.h>

typedef __attribute__((ext_vector_type(2))) float v2f;
typedef __attribute__((ext_vector_type(8))) float v8f;

#define D_LAT   64
#define KCLUS   64
#define N_ITERS 10
#define ALPHA_F 0.001f
#define EPS_F   1e-8f
#define NSPREAD 4            // interleaved accumulator copies (atomic de-contention)
#define RPB     128          // rows per block (8 waves x 16 rows) -- no-spill config

// Workspace layout (floats):
#define WS_DECLOSS 0
#define WS_CLULOSS 1                              // slots [1 .. N_ITERS] (one per iter)
#define WS_C       16
#define WS_CNEW    (WS_C + 4096)                  // NSPREAD copies of 64x64
#define WS_RSUM    (WS_CNEW + NSPREAD * 4096)     // NSPREAD copies of 64

// ---------------------------------------------------------------------------
// init: C = enc[:64,:], zero all loss slots + accumulator copies
// ---------------------------------------------------------------------------
__global__ void init_kernel(const float* __restrict__ enc, float* __restrict__ ws) {
    int t = blockIdx.x * blockDim.x + threadIdx.x;
    int stride = blockDim.x * gridDim.x;
    for (int i = t; i <= N_ITERS; i += stride) ws[i] = 0.f;   // dec loss + 10 clu slots
    for (int i = t; i < KCLUS * D_LAT; i += stride) ws[WS_C + i] = enc[i];
    for (int i = t; i < NSPREAD * KCLUS * D_LAT; i += stride) ws[WS_CNEW + i] = 0.f;
    for (int i = t; i < NSPREAD * KCLUS; i += stride) ws[WS_RSUM + i] = 0.f;
}

// ---------------------------------------------------------------------------
// decoder loss: sum((X - D)^2), float4 streaming, block reduce, one atomic
// ---------------------------------------------------------------------------
__global__ void dec_loss_kernel(const float* __restrict__ X,
                                const float* __restrict__ D,
                                float* __restrict__ ws, long n4) {
    __shared__ float red[8];
    long tid = (long)blockIdx.x * blockDim.x + threadIdx.x;
    long nthreads = (long)gridDim.x * blockDim.x;
    float s = 0.f;
    for (long i = tid; i < n4; i += nthreads) {
        float4 x = ((const float4*)X)[i];
        float4 d = ((const float4*)D)[i];
        float a = x.x - d.x, b = x.y - d.y, c = x.z - d.z, e = x.w - d.w;
        s += a * a + b * b + c * c + e * e;
    }
    for (int off = 16; off; off >>= 1) s += __shfl_xor(s, off, 32);
    int lane = threadIdx.x & 31, wave = threadIdx.x >> 5;
    if (lane == 0) red[wave] = s;
    __syncthreads();
    if (threadIdx.x == 0) {
        float b = 0.f;
        for (int w = 0; w < 8; ++w) b += red[w];
        atomicAdd(&ws[WS_DECLOSS], b);
    }
}

// ---------------------------------------------------------------------------
// assignment pass: d2 via V_WMMA_F32_16X16X4_F32, softmax, per-iter loss slot,
// Cnew += r.T @ enc (WMMA), rsum += colsum(r).
// Block = 256 threads = 8 waves; each wave owns 16 rows; block owns 128 rows.
// ---------------------------------------------------------------------------
__global__ void __launch_bounds__(256)
assign_kernel(const float* __restrict__ enc, float* __restrict__ ws,
              float* __restrict__ lossSlot) {
    __shared__ float sC[KCLUS * D_LAT];   // 16 KB cluster centers
    __shared__ float sE[RPB * D_LAT];     // 32 KB enc tile
    __shared__ float sR[RPB * KCLUS];     // 32 KB responsibilities
    __shared__ float sc2[KCLUS];
    __shared__ float sx2[RPB];
    __shared__ float sLoss;

    const int cp = blockIdx.x & (NSPREAD - 1);
    float* Cnew = ws + WS_CNEW + cp * (KCLUS * D_LAT);
    float* rsum = ws + WS_RSUM + cp * KCLUS;

    const int t = threadIdx.x;
    const long block_row0 = (long)blockIdx.x * RPB;

    // stage C and enc tile (128-bit LDS stores)
    for (int i = t; i < KCLUS * D_LAT / 4; i += 256)
        ((float4*)sC)[i] = ((const float4*)(ws + WS_C))[i];
    for (int i = t; i < RPB * D_LAT / 4; i += 256)
        ((float4*)sE)[i] = ((const float4*)(enc + block_row0 * D_LAT))[i];
    if (t == 0) sLoss = 0.f;
    __syncthreads();

    // c2[k] = |C_k|^2 ; x2[r] = |enc_r|^2
    if (t < KCLUS) {
        float s = 0.f;
        for (int d = 0; d < D_LAT; ++d) { float v = sC[t * D_LAT + d]; s += v * v; }
        sc2[t] = s;
    } else if (t < KCLUS + RPB) {
        int r = t - KCLUS;
        float s = 0.f;
        for (int d = 0; d < D_LAT; ++d) { float v = sE[r * D_LAT + d]; s += v * v; }
        sx2[r] = s;
    }
    __syncthreads();

    const int wave = t >> 5, lane = t & 31;
    const int half = lane >> 4, lq = lane & 15;
    const int wrow0 = wave * 16;   // local row base of this wave's 16-row tile

    // A fragments: A[m=lq][k = 4*ch + 2*half + {0,1}]  (fp32 16x4 layout)
    v2f a[16];
    {
        const float* rp = &sE[(wrow0 + lq) * D_LAT];
        for (int ch = 0; ch < 16; ++ch) {
            int col = 4 * ch + 2 * half;
            a[ch].x = rp[col];
            a[ch].y = rp[col + 1];
        }
    }
    float x2v[8];
    for (int v = 0; v < 8; ++v) x2v[v] = sx2[wrow0 + v + 8 * half];

    // d2 tile: 16 rows x 64 clusters (4 column tiles of 16)
    float d2v[4][8];
    for (int ct = 0; ct < 4; ++ct) {
        v8f acc = {};
        const float* cp0 = &sC[(ct * 16 + lq) * D_LAT];
        for (int ch = 0; ch < 16; ++ch) {
            int col = 4 * ch + 2 * half;
            v2f b;
            b.x = cp0[col];
            b.y = cp0[col + 1];
            acc = __builtin_amdgcn_wmma_f32_16x16x4_f32(false, a[ch], false, b,
                                                        (short)0, acc, false, false);
        }
        float c2v = sc2[ct * 16 + lq];
        for (int v = 0; v < 8; ++v) {
            float d2 = x2v[v] + c2v - 2.f * acc[v];
            d2v[ct][v] = d2 > 0.f ? d2 : 0.f;
        }
    }

    // row softmax over 64 clusters (butterflies stay inside 16-lane halves)
    float lossAcc = 0.f;
    for (int v = 0; v < 8; ++v) {
        float mn = d2v[0][v];
        for (int ct = 1; ct < 4; ++ct) mn = fminf(mn, d2v[ct][v]);
        for (int off = 1; off < 16; off <<= 1) mn = fminf(mn, __shfl_xor(mn, off, 32));
        float e[4], Z = 0.f, num = 0.f;
        for (int ct = 0; ct < 4; ++ct) {
            e[ct] = __expf(mn - d2v[ct][v]);   // softmax(-d2): shift by -min(d2)
            Z += e[ct];
            num += e[ct] * d2v[ct][v];
        }
        for (int off = 1; off < 16; off <<= 1) {
            Z += __shfl_xor(Z, off, 32);
            num += __shfl_xor(num, off, 32);
        }
        float invZ = 1.f / Z;
        int m = v + 8 * half;
        for (int ct = 0; ct < 4; ++ct)
            sR[(wrow0 + m) * KCLUS + ct * 16 + lq] = e[ct] * invZ;
        if (lq == 0) lossAcc += num * invZ;    // one lane per row contributes
    }
    if (lq == 0) atomicAdd(&sLoss, lossAcc);
    __syncthreads();
    if (t == 0) atomicAdd(lossSlot, sLoss);

    // Cnew += r.T (64x128) @ encTile (128x64); 16 output 16x16 tiles, 2 per wave
    for (int tt = 0; tt < 2; ++tt) {
        int tile = wave * 2 + tt;
        int ct = tile >> 2;     // cluster tile
        int dt = tile & 3;      // latent-dim tile
        v8f acc = {};
        for (int kc = 0; kc < RPB / 4; ++kc) {
            int row = kc * 4 + 2 * half;
            v2f av, bv;
            av.x = sR[row * KCLUS + ct * 16 + lq];
            av.y = sR[(row + 1) * KCLUS + ct * 16 + lq];
            bv.x = sE[row * D_LAT + dt * 16 + lq];
            bv.y = sE[(row + 1) * D_LAT + dt * 16 + lq];
            acc = __builtin_amdgcn_wmma_f32_16x16x4_f32(false, av, false, bv,
                                                        (short)0, acc, false, false);
        }
        for (int v = 0; v < 8; ++v) {
            int k = ct * 16 + v + 8 * half;   // cluster index (D-matrix M)
            int d = dt * 16 + lq;             // latent dim   (D-matrix N)
            atomicAdd(&Cnew[k * D_LAT + d], acc[v]);
        }
    }

    // rsum[k] += sum over this block's RPB rows of r[row][k]
    if (t < KCLUS) {
        float s = 0.f;
        for (int r = 0; r < RPB; ++r) s += sR[r * KCLUS + t];
        atomicAdd(&rsum[t], s);
    }
}

// ---------------------------------------------------------------------------
// C = sum(CnewCopies) / (sum(rsumCopies) + eps); re-zero accumulator copies
// ---------------------------------------------------------------------------
__global__ void update_kernel(float* __restrict__ ws) {
    __shared__ float rs[KCLUS];
    int t = threadIdx.x;
    if (t < KCLUS) {
        float s = 0.f;
        for (int c = 0; c < NSPREAD; ++c) s += ws[WS_RSUM + c * KCLUS + t];
        rs[t] = s + EPS_F;
    }
    __syncthreads();
    for (int i = t; i < KCLUS * D_LAT; i += 256) {
        float s = 0.f;
        for (int c = 0; c < NSPREAD; ++c) {
            s += ws[WS_CNEW + c * (KCLUS * D_LAT) + i];
            ws[WS_CNEW + c * (KCLUS * D_LAT) + i] = 0.f;
        }
        ws[WS_C + i] = s / rs[i >> 6];
    }
    if (t < NSPREAD * KCLUS) ws[WS_RSUM + t] = 0.f;
}

// ---------------------------------------------------------------------------
// out = decSum / (N*512) + ALPHA * cluSum_lastIter / N
// ---------------------------------------------------------------------------
__global__ void finalize_kernel(const float* __restrict__ ws, float* __restrict__ out,
                                float inv_nd, float inv_n) {
    if (threadIdx.x == 0)
        out[0] = ws[WS_DECLOSS] * inv_nd
               + ALPHA_F * ws[WS_CLULOSS + N_ITERS - 1] * inv_n;
}

// ---------------------------------------------------------------------------
extern "C" void kernel_launch(void* const* d_in, const int* in_sizes, int n_in,
                              void* d_out, int out_size, void* d_ws, size_t ws_size,
                              hipStream_t stream) {
    const float* X   = (const float*)d_in[0];
    const float* enc = (const float*)d_in[1];
    const float* dec = (const float*)d_in[2];
    float* out = (float*)d_out;
    float* ws  = (float*)d_ws;

    const long n_data = (long)in_sizes[0];          // N * 512
    const int  N      = in_sizes[1] / D_LAT;        // 65536

    init_kernel<<<64, 256, 0, stream>>>(enc, ws);
    dec_loss_kernel<<<4096, 256, 0, stream>>>(X, dec, ws, n_data / 4);

    const int assign_blocks = N / RPB;
    for (int it = 0; it < N_ITERS; ++it) {
        assign_kernel<<<assign_blocks, 256, 0, stream>>>(enc, ws,
                                                         ws + WS_CLULOSS + it);
        update_kernel<<<1, 256, 0, stream>>>(ws);
    }

    finalize_kernel<<<1, 32, 0, stream>>>(ws, out,
                                          1.f / (float)n_data,
                                          1.f / (float)N);
}